// PointTransformerEncoder_7765300871313
// MI455X (gfx1250) — compile-verified
//
#include <hip/hip_runtime.h>
#include <hip/hip_bf16.h>
#include <math.h>

// ---------------------------------------------------------------------------
// PointTransformer encoder for MI455X (gfx1250, wave32).
// All linear layers run through V_WMMA_F32_16X16X4_F32 (full fp32 semantics,
// matrix pipe). Workload is L2-resident (<<192MB), so the optimization target
// is matrix-pipe utilization + low per-iteration overhead, not HBM traffic.
// ---------------------------------------------------------------------------

typedef float v2f __attribute__((ext_vector_type(2)));
typedef float v8f __attribute__((ext_vector_type(8)));

#define BN_SCALE_F 0.9999950000374997f  // 1/sqrt(1+1e-5), eval-mode BN
#define CEILDIV(a, b) (((a) + (b) - 1) / (b))

static __device__ __forceinline__ v8f wmma4(v2f a, v2f b, v8f c) {
  return __builtin_amdgcn_wmma_f32_16x16x4_f32(
      /*neg_a=*/false, a, /*neg_b=*/false, b,
      /*c_mod=*/(short)0, c, /*reuse_a=*/false, /*reuse_b=*/false);
}

// ---------------------------------------------------------------------------
// Generic GEMM: out[M,N] = act( (A[M,K] @ W[K,N] + bias) * (g*BN_SCALE) + b )
// 128 threads = 4 waves per block; each wave owns one 16x16 output tile
// (4 consecutive M-tiles per block). Fast path: no predication, vector A
// loads, K unrolled x2. Fallback: guarded loop for edge tiles / odd K.
// Fragment layout per CDNA5 ISA 7.12.2 (32-bit A 16x4 / C 16x16 tables).
// ---------------------------------------------------------------------------
__global__ __launch_bounds__(128) void gemm_wmma_kernel(
    const float* __restrict__ A, const float* __restrict__ W,
    const float* __restrict__ bias, const float* __restrict__ bng,
    const float* __restrict__ bnb, float* __restrict__ out,
    int M, int K, int N, int relu) {
  const int wave = threadIdx.x >> 5;
  const int lane = threadIdx.x & 31;
  const int tm = (blockIdx.x * 4 + wave) * 16;
  const int tn = blockIdx.y * 16;
  if (tm >= M) return;

  const int half = lane >> 4;   // 0: K pair {0,1}, 1: K pair {2,3}
  const int l15  = lane & 15;
  const int arow = tm + l15;    // A row for this lane
  const int bcol = tn + l15;    // B/C column for this lane

  v8f acc = {};
  const bool full = (tm + 16 <= M) && (tn + 16 <= N);

  if (full && (K & 3) == 0) {
    // ---- fast path: fully in-range tile, K multiple of 4 ----
    const float* Ap = A + (size_t)arow * K + 2 * half;   // 8B aligned (K%4==0)
    const float* Wp = W + (size_t)(2 * half) * N + bcol;
    int k0 = 0;
    for (; k0 + 8 <= K; k0 += 8) {
      __builtin_prefetch(Ap + k0 + 32, 0, 0);            // global_prefetch_b8
      const v2f a0 = *(const v2f*)(Ap + k0);
      const v2f a1 = *(const v2f*)(Ap + k0 + 4);
      v2f b0, b1;
      b0.x = Wp[(size_t)k0 * N];
      b0.y = Wp[(size_t)k0 * N + N];
      b1.x = Wp[(size_t)(k0 + 4) * N];
      b1.y = Wp[(size_t)(k0 + 4) * N + N];
      acc = wmma4(a0, b0, acc);
      acc = wmma4(a1, b1, acc);
    }
    for (; k0 < K; k0 += 4) {
      const v2f a = *(const v2f*)(Ap + k0);
      v2f b;
      b.x = Wp[(size_t)k0 * N];
      b.y = Wp[(size_t)k0 * N + N];
      acc = wmma4(a, b, acc);
    }
  } else {
    // ---- guarded path: edge tiles and/or K not a multiple of 4 ----
    for (int k0 = 0; k0 < K; k0 += 4) {
      const int ka = k0 + 2 * half;
      v2f a, b;
      a.x = (arow < M && (ka    ) < K) ? A[(size_t)arow * K + ka    ] : 0.0f;
      a.y = (arow < M && (ka + 1) < K) ? A[(size_t)arow * K + ka + 1] : 0.0f;
      b.x = (bcol < N && (ka    ) < K) ? W[(size_t)(ka    ) * N + bcol] : 0.0f;
      b.y = (bcol < N && (ka + 1) < K) ? W[(size_t)(ka + 1) * N + bcol] : 0.0f;
      acc = wmma4(a, b, acc);
    }
  }

  // ---- epilogue: per-lane column is fixed; hoist bias/BN loads ----
  const bool colok = bcol < N;
  float add0 = 0.0f, scale = 1.0f, shift = 0.0f;
  if (colok) {
    if (bias) add0 = bias[bcol];
    if (bng) { scale = bng[bcol] * BN_SCALE_F; shift = bnb[bcol]; }
  }
#pragma unroll
  for (int i = 0; i < 8; ++i) {
    const int row = tm + i + 8 * half;   // C VGPR i -> row i / i+8 per half
    if (row < M && colok) {
      float v = (acc[i] + add0) * scale + shift;
      if (relu) v = v > 0.0f ? v : 0.0f;
      out[(size_t)row * N + bcol] = v;
    }
  }
}

// ---------------------------------------------------------------------------
// Furthest point sampling: one workgroup, LDS argmax reduction per iteration.
// Matches reference: idx[0]=0, iterative min-dist update + argmax (first-max).
// ---------------------------------------------------------------------------
__global__ void fps_kernel(const float* __restrict__ p, int* __restrict__ outidx,
                           float* __restrict__ dists, int n, int m) {
  __shared__ float sd[256];
  __shared__ int   si[256];
  __shared__ int   s_last;
  const int t = threadIdx.x;
  if (t == 0) { outidx[0] = 0; s_last = 0; }
  for (int j = t; j < n; j += 256) dists[j] = 1e10f;
  __syncthreads();
  for (int i = 1; i < m; ++i) {
    const int last = s_last;
    const float lx = p[3 * last], ly = p[3 * last + 1], lz = p[3 * last + 2];
    float best = -1e30f; int bi = 0x7fffffff;
    for (int j = t; j < n; j += 256) {
      const float dx = p[3 * j] - lx, dy = p[3 * j + 1] - ly, dz = p[3 * j + 2] - lz;
      float d = dx * dx + dy * dy + dz * dz;
      const float od = dists[j];
      d = d < od ? d : od;
      dists[j] = d;
      if (d > best || (d == best && j < bi)) { best = d; bi = j; }
    }
    sd[t] = best; si[t] = bi;
    __syncthreads();
    for (int off = 128; off > 0; off >>= 1) {
      if (t < off) {
        const float o = sd[t + off]; const int oi = si[t + off];
        if (o > sd[t] || (o == sd[t] && oi < si[t])) { sd[t] = o; si[t] = oi; }
      }
      __syncthreads();
    }
    if (t == 0) { outidx[i] = si[0]; s_last = si[0]; }
    __syncthreads();
  }
}

// ---------------------------------------------------------------------------
// Brute-force kNN: one thread per query, k-insertion (k<=16), ascending dist,
// ties keep lower index (matches lax.top_k stability).
// ---------------------------------------------------------------------------
__global__ void knn_kernel(const float* __restrict__ q, const float* __restrict__ r,
                           int* __restrict__ idx, int nq, int nr, int k) {
  const int i = blockIdx.x * blockDim.x + threadIdx.x;
  if (i >= nq) return;
  float bd[16]; int bi[16];
  for (int t = 0; t < 16; ++t) { bd[t] = 3.4e38f; bi[t] = 0; }
  const float qx = q[3 * i], qy = q[3 * i + 1], qz = q[3 * i + 2];
  const float qq = qx * qx + qy * qy + qz * qz;
  for (int j = 0; j < nr; ++j) {
    const float rx = r[3 * j], ry = r[3 * j + 1], rz = r[3 * j + 2];
    const float d = qq + rx * rx + ry * ry + rz * rz
                  - 2.0f * (qx * rx + qy * ry + qz * rz);
    if (d < bd[k - 1]) {
      int pos = k - 1;
      while (pos > 0 && d < bd[pos - 1]) {
        bd[pos] = bd[pos - 1]; bi[pos] = bi[pos - 1]; --pos;
      }
      bd[pos] = d; bi[pos] = j;
    }
  }
  for (int t = 0; t < k; ++t) idx[(size_t)i * k + t] = bi[t];
}

// ---------------------------------------------------------------------------
// Small elementwise / gather kernels
// ---------------------------------------------------------------------------
__global__ void gather_rows_kernel(const float* __restrict__ src, const int* __restrict__ idx,
                                   float* __restrict__ dst, int m, int d) {
  const int g = blockIdx.x * blockDim.x + threadIdx.x;
  if (g >= m * d) return;
  const int row = g / d, ch = g % d;
  dst[g] = src[(size_t)idx[row] * d + ch];
}

__global__ void build_g_kernel(const float* __restrict__ p, const float* __restrict__ x,
                               const float* __restrict__ newp, const int* __restrict__ idx,
                               float* __restrict__ g, int rows, int ns, int c) {
  const int d = 3 + c;
  const long gid = (long)blockIdx.x * blockDim.x + threadIdx.x;
  if (gid >= (long)rows * ns * d) return;
  const int row = (int)(gid / d), ch = (int)(gid % d);
  const int i = row / ns;
  const int src = idx[row];
  g[gid] = (ch < 3) ? (p[(size_t)src * 3 + ch] - newp[(size_t)i * 3 + ch])
                    : x[(size_t)src * c + (ch - 3)];
}

__global__ void max_ns_kernel(const float* __restrict__ y, float* __restrict__ out,
                              int m, int ns, int c) {
  const int g = blockIdx.x * blockDim.x + threadIdx.x;
  if (g >= m * c) return;
  const int i = g / c, ch = g % c;
  float v = -3.4e38f;
  for (int j = 0; j < ns; ++j) {
    const float t = y[((size_t)i * ns + j) * c + ch];
    v = t > v ? t : v;
  }
  out[g] = v;
}

__global__ void gather_pr_kernel(const float* __restrict__ p, const int* __restrict__ idx,
                                 float* __restrict__ pr, int n, int ns) {
  const int g = blockIdx.x * blockDim.x + threadIdx.x;
  if (g >= n * ns * 3) return;
  const int row = g / 3, ch = g % 3;
  const int i = row / ns;
  pr[g] = p[(size_t)idx[row] * 3 + ch] - p[(size_t)i * 3 + ch];
}

__global__ void combine_w_kernel(const float* __restrict__ kmat, const float* __restrict__ q,
                                 const float* __restrict__ prc, const int* __restrict__ idx,
                                 float* __restrict__ w, int n, int ns, int c) {
  const long gid = (long)blockIdx.x * blockDim.x + threadIdx.x;
  if (gid >= (long)n * ns * c) return;
  const int row = (int)(gid / c), ch = (int)(gid % c);
  const int i = row / ns;
  w[gid] = kmat[(size_t)idx[row] * c + ch] - q[(size_t)i * c + ch] + prc[gid];
}

__global__ void bn_act_kernel(const float* __restrict__ in, float* __restrict__ out,
                              const float* __restrict__ g, const float* __restrict__ b,
                              long rows, int c, int relu) {
  const long gid = (long)blockIdx.x * blockDim.x + threadIdx.x;
  if (gid >= rows * c) return;
  const int ch = (int)(gid % c);
  float v = in[gid] * (g[ch] * BN_SCALE_F) + b[ch];
  if (relu) v = v > 0.0f ? v : 0.0f;
  out[gid] = v;
}

__global__ void softmax_ns_kernel(float* __restrict__ w, int n, int ns, int cs) {
  const int g = blockIdx.x * blockDim.x + threadIdx.x;
  if (g >= n * cs) return;
  const int i = g / cs, ch = g % cs;
  float mx = -3.4e38f;
  for (int j = 0; j < ns; ++j) {
    const float v = w[((size_t)i * ns + j) * cs + ch];
    mx = v > mx ? v : mx;
  }
  float ev[16], s = 0.0f;
  for (int j = 0; j < ns; ++j) {
    ev[j] = expf(w[((size_t)i * ns + j) * cs + ch] - mx);
    s += ev[j];
  }
  const float inv = 1.0f / s;
  for (int j = 0; j < ns; ++j)
    w[((size_t)i * ns + j) * cs + ch] = ev[j] * inv;
}

__global__ void attn_out_kernel(const float* __restrict__ v, const float* __restrict__ prc,
                                const float* __restrict__ w, const int* __restrict__ idx,
                                float* __restrict__ out, int n, int ns, int c, int cs) {
  const int g = blockIdx.x * blockDim.x + threadIdx.x;
  if (g >= n * c) return;
  const int i = g / c, ch = g % c;
  const int within = ch % cs;  // reshape(n,ns,SHARE,c/SHARE): share-group major
  float acc = 0.0f;
  for (int j = 0; j < ns; ++j) {
    const size_t row = (size_t)i * ns + j;
    acc += (v[(size_t)idx[row] * c + ch] + prc[row * c + ch]) * w[row * cs + within];
  }
  out[g] = acc;
}

__global__ void add_relu_kernel(const float* __restrict__ a, const float* __restrict__ b,
                                float* __restrict__ out, long n) {
  const long gid = (long)blockIdx.x * blockDim.x + threadIdx.x;
  if (gid >= n) return;
  const float v = a[gid] + b[gid];
  out[gid] = v > 0.0f ? v : 0.0f;
}

// ---------------------------------------------------------------------------
// Host orchestration
// ---------------------------------------------------------------------------
namespace {

constexpr int B_ = 2, N_ = 8192, IN_C_ = 6, SHARE_ = 8;
constexpr int PLANES_[5] = {32, 64, 128, 256, 512};
constexpr int BLOCKS_[5] = {2, 3, 4, 6, 3};
constexpr int STRIDE_[5] = {1, 4, 4, 4, 4};
constexpr int NSAMP_[5]  = {8, 16, 16, 16, 16};

// jax pytree (key-sorted) leaf offsets within one attention block (28 leaves)
enum {
  L_BN1_B = 0, L_BN1_G, L_BN2_B, L_BN2_G, L_BN3_B, L_BN3_G,
  L_K_B, L_K_W, L_LIN1_W, L_LIN3_W,
  L_P1_B, L_P1_W, L_P2_B, L_P2_W, L_PBN_B, L_PBN_G,
  L_Q_B, L_Q_W, L_V_B, L_V_W,
  L_W1_B, L_W1_W, L_W2_B, L_W2_W,
  L_WBN1_B, L_WBN1_G, L_WBN2_B, L_WBN2_G,
  LEAVES_PER_BLOCK = 28
};

inline int stage_base_leaf(int s) {
  int b = 0;
  for (int i = 0; i < s; ++i) b += (BLOCKS_[i] - 1) * LEAVES_PER_BLOCK + 3;
  return b;
}

inline void gemm(hipStream_t st, const float* A, const float* W, const float* bias,
                 const float* g, const float* b, float* out,
                 int M, int K, int N, int relu) {
  dim3 grid(CEILDIV(M, 64), CEILDIV(N, 16));  // 4 waves/block, 16 rows each
  gemm_wmma_kernel<<<grid, dim3(128), 0, st>>>(A, W, bias, g, b, out, M, K, N, relu);
}

}  // namespace

extern "C" void kernel_launch(void* const* d_in, const int* in_sizes, int n_in,
                              void* d_out, int out_size, void* d_ws, size_t ws_size,
                              hipStream_t stream) {
  (void)in_sizes; (void)n_in; (void)out_size; (void)ws_size;

  const float* points = (const float*)d_in[0];  // [B,N,3]
  const float* feats  = (const float*)d_in[1];  // [B,N,6]
  // d_in[2] = offset (unused: equal-sized clouds), params leaves start at [3]
  auto leaf = [&](int i) -> const float* { return (const float*)d_in[3 + i]; };

  // ---- bump allocator on d_ws ----
  char* wsbase = (char*)d_ws;
  size_t wsoff = 0;
  auto allocf = [&](size_t nelem) -> float* {
    float* p = (float*)(wsbase + wsoff);
    wsoff += ((nelem * sizeof(float) + 255) / 256) * 256;
    return p;
  };
  auto alloci = [&](size_t nelem) -> int* {
    int* p = (int*)(wsbase + wsoff);
    wsoff += ((nelem * sizeof(int) + 255) / 256) * 256;
    return p;
  };

  const size_t MAXNC    = 262144;   // max n*c per batch (8192*32)
  const size_t MAXNNS_C = 2097152;  // max n*ns*c per batch
  const size_t MAXNNS3  = 196608;   // max n*ns*3 per batch
  const size_t MAXNNSCS = 262144;   // max n*ns*(c/8) per batch
  const size_t MAXG     = 1146880;  // max m*ns*(3+c_in)
  const size_t MAXTDY   = 2097152;  // max m*ns*c

  float* Pbuf[2] = {allocf((size_t)B_ * N_ * 3), allocf((size_t)B_ * N_ * 3)};
  float* Xbuf[2] = {allocf((size_t)B_ * MAXNC), allocf((size_t)B_ * MAXNC)};
  int*   idx_td  = alloci(2048 * 16);
  int*   idx_at  = alloci((size_t)N_ * 16);
  int*   fpsi    = alloci(2048);
  float* fpsd    = allocf(N_);
  float* Qb = allocf(MAXNC); float* Kb = allocf(MAXNC); float* Vb = allocf(MAXNC);
  float* PR3  = allocf(MAXNNS3);
  float* PR3b = allocf(MAXNNS3);
  float* PRC  = allocf(MAXNNS_C);
  float* WBUF = allocf(MAXNNS_C);
  float* WT1  = allocf(MAXNNSCS);
  float* WT2  = allocf(MAXNNSCS);
  float* Y1 = allocf(MAXNC); float* Y2 = allocf(MAXNC);
  float* Gbuf = allocf(MAXG);
  float* TDY  = allocf(MAXTDY);

  const float* curP = points;  // per-batch stride cur_n*3
  const float* curX = feats;   // per-batch stride cur_n*cur_c
  int cur_n = N_, cur_c = IN_C_;
  int ppick = 0, xpick = 0;

  float* outf = (float*)d_out;
  size_t ooff = 0;

  for (int s = 0; s < 5; ++s) {
    const int c    = PLANES_[s];
    const int ns   = NSAMP_[s];
    const int nblk = BLOCKS_[s] - 1;
    const int cs   = c / SHARE_;
    const int sb   = stage_base_leaf(s);
    const int tdb  = sb + nblk * LEAVES_PER_BLOCK;
    const float* td_bn_b = leaf(tdb + 0);
    const float* td_bn_g = leaf(tdb + 1);
    const float* td_w    = leaf(tdb + 2);

    const int n_in = cur_n, c_in = cur_c;
    const int n = (STRIDE_[s] == 1) ? n_in : n_in / 4;
    float* newP = Pbuf[ppick];
    float* newX = Xbuf[xpick];

    for (int b = 0; b < B_; ++b) {
      const float* pin = curP + (size_t)b * n_in * 3;
      const float* xin = curX + (size_t)b * n_in * c_in;
      float* pout = newP + (size_t)b * n * 3;
      float* xout = newX + (size_t)b * n * c;

      // ---- transition down ----
      if (STRIDE_[s] == 1) {
        gemm(stream, xin, td_w, nullptr, td_bn_g, td_bn_b, xout, n_in, c_in, c, 1);
        hipMemcpyAsync(pout, pin, (size_t)n_in * 3 * sizeof(float),
                       hipMemcpyDeviceToDevice, stream);
      } else {
        fps_kernel<<<1, 256, 0, stream>>>(pin, fpsi, fpsd, n_in, n);
        gather_rows_kernel<<<CEILDIV(n * 3, 256), 256, 0, stream>>>(pin, fpsi, pout, n, 3);
        knn_kernel<<<CEILDIV(n, 128), 128, 0, stream>>>(pout, pin, idx_td, n, n_in, ns);
        const long grows = (long)n * ns * (3 + c_in);
        build_g_kernel<<<CEILDIV(grows, 256L), 256, 0, stream>>>(
            pin, xin, pout, idx_td, Gbuf, n, ns, c_in);
        gemm(stream, Gbuf, td_w, nullptr, td_bn_g, td_bn_b, TDY, n * ns, 3 + c_in, c, 1);
        max_ns_kernel<<<CEILDIV(n * c, 256), 256, 0, stream>>>(TDY, xout, n, ns, c);
      }

      // ---- attention kNN (p is fixed for the whole stage) ----
      if (nblk > 0)
        knn_kernel<<<CEILDIV(n, 128), 128, 0, stream>>>(pout, pout, idx_at, n, n, ns);

      // ---- PointTransformer blocks ----
      for (int k = 0; k < nblk; ++k) {
        const int bb = sb + k * LEAVES_PER_BLOCK;
        const float* bn1_b = leaf(bb + L_BN1_B); const float* bn1_g = leaf(bb + L_BN1_G);
        const float* bn2_b = leaf(bb + L_BN2_B); const float* bn2_g = leaf(bb + L_BN2_G);
        const float* bn3_b = leaf(bb + L_BN3_B); const float* bn3_g = leaf(bb + L_BN3_G);
        const float* k_b   = leaf(bb + L_K_B);   const float* k_w   = leaf(bb + L_K_W);
        const float* lin1w = leaf(bb + L_LIN1_W);
        const float* lin3w = leaf(bb + L_LIN3_W);
        const float* p1_b  = leaf(bb + L_P1_B);  const float* p1_w  = leaf(bb + L_P1_W);
        const float* p2_b  = leaf(bb + L_P2_B);  const float* p2_w  = leaf(bb + L_P2_W);
        const float* pbn_b = leaf(bb + L_PBN_B); const float* pbn_g = leaf(bb + L_PBN_G);
        const float* q_b   = leaf(bb + L_Q_B);   const float* q_w   = leaf(bb + L_Q_W);
        const float* v_b   = leaf(bb + L_V_B);   const float* v_w   = leaf(bb + L_V_W);
        const float* w1_b  = leaf(bb + L_W1_B);  const float* w1_w  = leaf(bb + L_W1_W);
        const float* w2_b  = leaf(bb + L_W2_B);  const float* w2_w  = leaf(bb + L_W2_W);
        const float* wbn1b = leaf(bb + L_WBN1_B); const float* wbn1g = leaf(bb + L_WBN1_G);
        const float* wbn2b = leaf(bb + L_WBN2_B); const float* wbn2g = leaf(bb + L_WBN2_G);

        const long nnsc  = (long)n * ns * c;
        const long nnscs = (long)n * ns * cs;

        // y = relu(bn1(lin1(x)))
        gemm(stream, xout, lin1w, nullptr, bn1_g, bn1_b, Y1, n, c, c, 1);

        // q, k, v projections
        gemm(stream, Y1, q_w, q_b, nullptr, nullptr, Qb, n, c, c, 0);
        gemm(stream, Y1, k_w, k_b, nullptr, nullptr, Kb, n, c, c, 0);
        gemm(stream, Y1, v_w, v_b, nullptr, nullptr, Vb, n, c, c, 0);

        // positional branch: pr = p[idx]-p ; p2(relu(pbn(p1(pr))))
        gather_pr_kernel<<<CEILDIV(n * ns * 3, 256), 256, 0, stream>>>(pout, idx_at, PR3, n, ns);
        gemm(stream, PR3, p1_w, p1_b, pbn_g, pbn_b, PR3b, n * ns, 3, 3, 1);
        gemm(stream, PR3b, p2_w, p2_b, nullptr, nullptr, PRC, n * ns, 3, c, 0);

        // w = k[idx] - q + pr ; MLP ; softmax over ns
        combine_w_kernel<<<CEILDIV(nnsc, 256L), 256, 0, stream>>>(Kb, Qb, PRC, idx_at, WBUF, n, ns, c);
        bn_act_kernel<<<CEILDIV(nnsc, 256L), 256, 0, stream>>>(WBUF, WBUF, wbn1g, wbn1b, (long)n * ns, c, 1);
        gemm(stream, WBUF, w1_w, w1_b, nullptr, nullptr, WT1, n * ns, c, cs, 0);
        bn_act_kernel<<<CEILDIV(nnscs, 256L), 256, 0, stream>>>(WT1, WT1, wbn2g, wbn2b, (long)n * ns, cs, 1);
        gemm(stream, WT1, w2_w, w2_b, nullptr, nullptr, WT2, n * ns, cs, cs, 0);
        softmax_ns_kernel<<<CEILDIV(n * cs, 256), 256, 0, stream>>>(WT2, n, ns, cs);

        // out = sum_j (v[idx]+pr) * w  (share-grouped)
        attn_out_kernel<<<CEILDIV(n * c, 256), 256, 0, stream>>>(Vb, PRC, WT2, idx_at, Y2, n, ns, c, cs);

        // y = relu(bn2(out)); y = bn3(lin3(y)); x = relu(y + identity)
        bn_act_kernel<<<CEILDIV((long)n * c, 256L), 256, 0, stream>>>(Y2, Y2, bn2_g, bn2_b, n, c, 1);
        gemm(stream, Y2, lin3w, nullptr, bn3_g, bn3_b, Y1, n, c, c, 0);
        add_relu_kernel<<<CEILDIV((long)n * c, 256L), 256, 0, stream>>>(Y1, xout, xout, (long)n * c);
      }
    }

    // ---- stage outputs: p [B*n,3] then x [B*n,c] ----
    hipMemcpyAsync(outf + ooff, newP, (size_t)B_ * n * 3 * sizeof(float),
                   hipMemcpyDeviceToDevice, stream);
    ooff += (size_t)B_ * n * 3;
    hipMemcpyAsync(outf + ooff, newX, (size_t)B_ * n * c * sizeof(float),
                   hipMemcpyDeviceToDevice, stream);
    ooff += (size_t)B_ * n * c;

    curP = newP; curX = newX; cur_n = n; cur_c = c;
    ppick ^= 1; xpick ^= 1;
  }
}